// TPCL_25323127177287
// MI455X (gfx1250) — compile-verified
//
#include <hip/hip_runtime.h>

#define N_NODES 10000
#define N_EDGES 100000
#define HIDDEN 128
#define MUL0 32
#define MUL1 8
#define IN_DIM 56            // MUL0 + 3*MUL1
#define W_NUMEL 1600
#define EPS 1e-5f

typedef __attribute__((ext_vector_type(2))) float v2f;
typedef __attribute__((ext_vector_type(8))) float v8f;

// workspace layout (in floats)
#define WS_NUM   0                         // N_NODES*IN_DIM
#define WS_CNT   (N_NODES*IN_DIM)          // N_NODES
#define WS_STAT  (WS_CNT + N_NODES)        // 72: sum_s[32] | sumsq_s[32] | sumsq_v[8]
#define WS_TOTAL (WS_STAT + 72)

__global__ __launch_bounds__(256) void zero_ws_kernel(float* __restrict__ ws) {
  int i = blockIdx.x * 256 + threadIdx.x;
  if (i < WS_TOTAL) ws[i] = 0.0f;
}

// ---------------------------------------------------------------------------
// Fused edge kernel: 32 edges / block, 256 threads (8 waves).
//   GEMM1: h = relu(edge_attr(32x128) @ fc1_w(128x128) + b1)   [f32 WMMA]
//   GEMM2: w = h @ fc2_w(128x1600) + b2, consumed tile-by-tile [f32 WMMA]
//   Tensor product -> per-edge tp[56] in LDS -> atomic scatter to nodes.
// ---------------------------------------------------------------------------
__global__ __launch_bounds__(256)
void edge_kernel(const float* __restrict__ node_attr,
                 const int*   __restrict__ edge_index,
                 const float* __restrict__ edge_attr,
                 const float* __restrict__ edge_sh,
                 const float* __restrict__ fc1_w, const float* __restrict__ fc1_b,
                 const float* __restrict__ fc2_w, const float* __restrict__ fc2_b,
                 float* __restrict__ num, float* __restrict__ cnt)
{
  __shared__ float sA[32 * 132];      // edge_attr tile (padded stride)
  __shared__ float sH[32 * 132];      // hidden tile (padded stride)
  __shared__ float sX[32 * IN_DIM];   // node_attr[dst] per edge
  __shared__ float sY[32 * 4];        // edge_sh per edge
  __shared__ float sZ0[32 * MUL0];    // x0*y0
  __shared__ float sZ1[32 * MUL1];    // sum_m x1[u,m]*y1[m]
  __shared__ float sTP[32 * IN_DIM];  // tensor-product accumulator
  __shared__ int   sSrc[32];

  const int tid  = threadIdx.x;
  const int lane = tid & 31;
  const int wv   = tid >> 5;
  const int e0   = blockIdx.x * 32;

  // ---- stage inputs ----
  for (int i = tid; i < 32 * HIDDEN; i += 256) {
    int r = i >> 7, c = i & 127;
    sA[r * 132 + c] = edge_attr[(size_t)(e0 + r) * HIDDEN + c];
  }
  if (tid < 32) sSrc[tid] = edge_index[e0 + tid];
  if (tid < 128) {
    int r = tid >> 2, c = tid & 3;
    sY[r * 4 + c] = edge_sh[(size_t)(e0 + r) * 4 + c];
  }
  for (int i = tid; i < 32 * IN_DIM; i += 256) {
    int r = i / IN_DIM, c = i - r * IN_DIM;
    int dst = edge_index[N_EDGES + e0 + r];
    sX[i]  = node_attr[(size_t)dst * IN_DIM + c];
    sTP[i] = 0.0f;
  }
  __syncthreads();

  // ---- per-edge TP coefficients ----
  for (int i = tid; i < 32 * MUL0; i += 256) {
    int r = i >> 5, u = i & 31;
    sZ0[i] = sX[r * IN_DIM + u] * sY[r * 4 + 0];
  }
  for (int i = tid; i < 32 * MUL1; i += 256) {
    int r = i >> 3, u = i & 7;
    const float* xr = &sX[r * IN_DIM + MUL0 + u * 3];
    const float* yr = &sY[r * 4 + 1];
    sZ1[i] = xr[0] * yr[0] + xr[1] * yr[1] + xr[2] * yr[2];
  }
  __syncthreads();

  const int m     = lane & 15;          // A row within tile / B column
  const int koff  = (lane >> 4) << 1;   // K sub-offset {0,2}
  const int mrow0 = (lane >> 4) << 3;   // C row base {0,8}

  // ---- GEMM1: wave wv computes hidden columns [wv*16, wv*16+16) ----
  {
    v8f acc0 = {}, acc1 = {};
    const int col = wv * 16 + m;
    #pragma unroll 8
    for (int kb = 0; kb < HIDDEN; kb += 4) {
      v2f a0, a1, b;
      a0.x = sA[m * 132 + kb + koff];
      a0.y = sA[m * 132 + kb + koff + 1];
      a1.x = sA[(16 + m) * 132 + kb + koff];
      a1.y = sA[(16 + m) * 132 + kb + koff + 1];
      b.x  = fc1_w[(size_t)(kb + koff) * HIDDEN + col];
      b.y  = fc1_w[(size_t)(kb + koff + 1) * HIDDEN + col];
      acc0 = __builtin_amdgcn_wmma_f32_16x16x4_f32(false, a0, false, b, (short)0, acc0, false, false);
      acc1 = __builtin_amdgcn_wmma_f32_16x16x4_f32(false, a1, false, b, (short)0, acc1, false, false);
    }
    const float bias = fc1_b[col];
    #pragma unroll
    for (int v = 0; v < 8; ++v) {
      float h0 = acc0[v] + bias;
      float h1 = acc1[v] + bias;
      sH[(mrow0 + v) * 132 + col]      = h0 > 0.0f ? h0 : 0.0f;
      sH[(16 + mrow0 + v) * 132 + col] = h1 > 0.0f ? h1 : 0.0f;
    }
  }
  __syncthreads();

  // ---- GEMM2 + tensor-product consumption ----
  const float alpha = 0.15811388300841897f;             // 1/sqrt(MUL0+MUL1)
  const float ais3  = alpha * 0.5773502691896258f;      // alpha/sqrt(3)

  for (int nt = wv; nt < 100; nt += 8) {               // 1600/16 column tiles
    v8f cc[2] = {{}, {}};
    const int col = nt * 16 + m;                        // j in [0,1600)
    #pragma unroll 8
    for (int kb = 0; kb < HIDDEN; kb += 4) {
      v2f a0, a1, b;
      a0.x = sH[m * 132 + kb + koff];
      a0.y = sH[m * 132 + kb + koff + 1];
      a1.x = sH[(16 + m) * 132 + kb + koff];
      a1.y = sH[(16 + m) * 132 + kb + koff + 1];
      b.x  = fc2_w[(size_t)(kb + koff) * W_NUMEL + col];
      b.y  = fc2_w[(size_t)(kb + koff + 1) * W_NUMEL + col];
      cc[0] = __builtin_amdgcn_wmma_f32_16x16x4_f32(false, a0, false, b, (short)0, cc[0], false, false);
      cc[1] = __builtin_amdgcn_wmma_f32_16x16x4_f32(false, a1, false, b, (short)0, cc[1], false, false);
    }
    const float bias = fc2_b[col];

    // weight-block boundaries (1024/1280/1536) are multiples of 16 -> branch
    // is uniform per wave for every tile nt.
    #pragma unroll
    for (int t = 0; t < 2; ++t) {
      const int eb = t * 16;
      if (col < 1024) {                                 // w00_0[u,w] -> out0[w]
        const int u = col >> 5, w = col & 31;
        #pragma unroll
        for (int v = 0; v < 8; ++v) {
          int e = eb + mrow0 + v;
          float val = cc[t][v] + bias;
          unsafeAtomicAdd(&sTP[e * IN_DIM + w], alpha * sZ0[e * MUL0 + u] * val);
        }
      } else if (col < 1280) {                          // w11_0[u,w] -> out0[w]
        const int jj = col - 1024, u = jj >> 5, w = jj & 31;
        #pragma unroll
        for (int v = 0; v < 8; ++v) {
          int e = eb + mrow0 + v;
          float val = cc[t][v] + bias;
          unsafeAtomicAdd(&sTP[e * IN_DIM + w], ais3 * sZ1[e * MUL1 + u] * val);
        }
      } else if (col < 1536) {                          // w01_1[u,w] -> out1[w,m]
        const int jj = col - 1280, u = jj >> 3, w = jj & 7;
        #pragma unroll
        for (int v = 0; v < 8; ++v) {
          int e = eb + mrow0 + v;
          float val = alpha * (cc[t][v] + bias) * sX[e * IN_DIM + u];
          unsafeAtomicAdd(&sTP[e * IN_DIM + MUL0 + w * 3 + 0], val * sY[e * 4 + 1]);
          unsafeAtomicAdd(&sTP[e * IN_DIM + MUL0 + w * 3 + 1], val * sY[e * 4 + 2]);
          unsafeAtomicAdd(&sTP[e * IN_DIM + MUL0 + w * 3 + 2], val * sY[e * 4 + 3]);
        }
      } else {                                          // w10_1[u,w] -> out1[w,m]
        const int jj = col - 1536, u = jj >> 3, w = jj & 7;
        #pragma unroll
        for (int v = 0; v < 8; ++v) {
          int e = eb + mrow0 + v;
          float val = alpha * (cc[t][v] + bias) * sY[e * 4 + 0];
          unsafeAtomicAdd(&sTP[e * IN_DIM + MUL0 + w * 3 + 0], val * sX[e * IN_DIM + MUL0 + u * 3 + 0]);
          unsafeAtomicAdd(&sTP[e * IN_DIM + MUL0 + w * 3 + 1], val * sX[e * IN_DIM + MUL0 + u * 3 + 1]);
          unsafeAtomicAdd(&sTP[e * IN_DIM + MUL0 + w * 3 + 2], val * sX[e * IN_DIM + MUL0 + u * 3 + 2]);
        }
      }
    }
  }
  __syncthreads();

  // ---- scatter to node accumulators ----
  for (int i = tid; i < 32 * IN_DIM; i += 256) {
    int r = i / IN_DIM, c = i - r * IN_DIM;
    unsafeAtomicAdd(&num[(size_t)sSrc[r] * IN_DIM + c], sTP[i]);
  }
  if (tid < 32) unsafeAtomicAdd(&cnt[sSrc[tid]], 1.0f);
}

// ---------------------------------------------------------------------------
// Node pass 1: out_pre = num/max(cnt,1) + node_attr (written to d_out),
// accumulate batch statistics into stat[72].
// ---------------------------------------------------------------------------
__global__ __launch_bounds__(128)
void node_pass1(const float* __restrict__ node_attr,
                const float* __restrict__ num, const float* __restrict__ cnt,
                float* __restrict__ outp, float* __restrict__ stat)
{
  __shared__ float ssum[MUL0], ssq[MUL0], svq[MUL1];
  const int tid = threadIdx.x;
  if (tid < MUL0) { ssum[tid] = 0.0f; ssq[tid] = 0.0f; }
  if (tid < MUL1) svq[tid] = 0.0f;
  __syncthreads();

  const int n = blockIdx.x * 128 + tid;
  if (n < N_NODES) {
    float c   = cnt[n];
    float inv = 1.0f / (c > 1.0f ? c : 1.0f);
    for (int ch = 0; ch < MUL0; ++ch) {
      float p = num[(size_t)n * IN_DIM + ch] * inv + node_attr[(size_t)n * IN_DIM + ch];
      outp[(size_t)n * IN_DIM + ch] = p;
      unsafeAtomicAdd(&ssum[ch], p);
      unsafeAtomicAdd(&ssq[ch], p * p);
    }
    for (int u = 0; u < MUL1; ++u) {
      float acc = 0.0f;
      for (int mm = 0; mm < 3; ++mm) {
        int ch = MUL0 + u * 3 + mm;
        float p = num[(size_t)n * IN_DIM + ch] * inv + node_attr[(size_t)n * IN_DIM + ch];
        outp[(size_t)n * IN_DIM + ch] = p;
        acc += p * p;
      }
      unsafeAtomicAdd(&svq[u], acc);
    }
  }
  __syncthreads();
  if (tid < MUL0) {
    unsafeAtomicAdd(&stat[tid], ssum[tid]);
    unsafeAtomicAdd(&stat[MUL0 + tid], ssq[tid]);
  }
  if (tid < MUL1) unsafeAtomicAdd(&stat[2 * MUL0 + tid], svq[tid]);
}

// ---------------------------------------------------------------------------
// Node pass 2: in-place batch normalization of d_out using stat[72].
// ---------------------------------------------------------------------------
__global__ __launch_bounds__(128)
void node_pass2(const float* __restrict__ stat,
                const float* __restrict__ bn_w_s, const float* __restrict__ bn_w_v,
                const float* __restrict__ bn_b_s,
                float* __restrict__ out)
{
  const int n = blockIdx.x * 128 + threadIdx.x;
  if (n >= N_NODES) return;
  const float invN = 1.0f / (float)N_NODES;
  for (int ch = 0; ch < MUL0; ++ch) {
    float mean = stat[ch] * invN;
    float var  = stat[MUL0 + ch] * invN - mean * mean;   // E[s^2]-E[s]^2 = E[(s-E[s])^2]
    float p = out[(size_t)n * IN_DIM + ch];
    out[(size_t)n * IN_DIM + ch] = (p - mean) * rsqrtf(var + EPS) * bn_w_s[ch] + bn_b_s[ch];
  }
  for (int u = 0; u < MUL1; ++u) {
    float vn    = stat[2 * MUL0 + u] * invN * (1.0f / 3.0f);
    float scale = rsqrtf(vn + EPS) * bn_w_v[u];
    for (int mm = 0; mm < 3; ++mm) {
      int ch = MUL0 + u * 3 + mm;
      out[(size_t)n * IN_DIM + ch] *= scale;
    }
  }
}

extern "C" void kernel_launch(void* const* d_in, const int* in_sizes, int n_in,
                              void* d_out, int out_size, void* d_ws, size_t ws_size,
                              hipStream_t stream) {
  const float* node_attr = (const float*)d_in[0];
  const int*   edge_index = (const int*)d_in[1];
  const float* edge_attr = (const float*)d_in[2];
  const float* edge_sh   = (const float*)d_in[3];
  const float* fc1_w = (const float*)d_in[4];
  const float* fc1_b = (const float*)d_in[5];
  const float* fc2_w = (const float*)d_in[6];
  const float* fc2_b = (const float*)d_in[7];
  const float* bn_w_s = (const float*)d_in[8];
  const float* bn_w_v = (const float*)d_in[9];
  const float* bn_b_s = (const float*)d_in[10];

  float* ws   = (float*)d_ws;
  float* num  = ws + WS_NUM;
  float* cnt  = ws + WS_CNT;
  float* stat = ws + WS_STAT;
  float* out  = (float*)d_out;

  zero_ws_kernel<<<(WS_TOTAL + 255) / 256, 256, 0, stream>>>(ws);
  edge_kernel<<<N_EDGES / 32, 256, 0, stream>>>(node_attr, edge_index, edge_attr, edge_sh,
                                                fc1_w, fc1_b, fc2_w, fc2_b, num, cnt);
  node_pass1<<<(N_NODES + 127) / 128, 128, 0, stream>>>(node_attr, num, cnt, out, stat);
  node_pass2<<<(N_NODES + 127) / 128, 128, 0, stream>>>(stat, bn_w_s, bn_w_v, bn_b_s, out);
}